// WHSys_16939351015995
// MI455X (gfx1250) — compile-verified
//
#include <hip/hip_runtime.h>
#include <hip/hip_bf16.h>
#include <math.h>

// Problem constants (match reference)
static constexpr int Bsz = 512;
static constexpr int T   = 16384;
static constexpr int L   = 512;      // chunk length for parallel scan
static constexpr int C   = T / L;    // 32 chunks per row

typedef __attribute__((ext_vector_type(2))) float v2f;
typedef __attribute__((ext_vector_type(4))) float v4f;
typedef __attribute__((ext_vector_type(8))) float v8f;

// y = -elu(-10/11 * x)  ==  x>=0 ? 1-exp(-c x) : c x
// Branchless: compute both sides, sign-bit select. v_exp_f32 is a TRANS op
// that co-executes with VALU; avoids the EXEC-mask diamonds clang otherwise
// emits per element.
__device__ __forceinline__ float whf(float x) {
    const float c   = 10.0f / 11.0f;
    const float l2e = 1.4426950408889634f;          // log2(e)
    float e = 1.0f - __builtin_amdgcn_exp2f(-(c * l2e) * x);
    float l = c * x;
    unsigned s = (unsigned)(__float_as_int(x) >> 31);  // all-ones if x < 0
    return __uint_as_float((__float_as_uint(e) & ~s) | (__float_as_uint(l) & s));
}

// ---------------------------------------------------------------------------
// K0: homogeneous impulse responses h_j[i] (response to initial state e_j,
// zero input) packed as hp[i*4 + j], j=3 column is zero (K=4 pad for WMMA).
// block 0 -> filter 1, block 1 -> filter 2.
// ---------------------------------------------------------------------------
__global__ void hpack_kernel(const float* __restrict__ a1c,
                             const float* __restrict__ a2c,
                             float* __restrict__ hp1,
                             float* __restrict__ hp2) {
    const float* ac = (blockIdx.x == 0) ? a1c : a2c;
    float*       hp = (blockIdx.x == 0) ? hp1 : hp2;
    int j = threadIdx.x;
    if (j > 3) return;
    if (j == 3) {
        for (int i = 0; i < L; ++i) hp[i * 4 + 3] = 0.0f;
        return;
    }
    const float a0 = ac[0], a1 = ac[1], a2 = ac[2];
    float y1 = (j == 0) ? 1.0f : 0.0f;   // y[-1]
    float y2 = (j == 1) ? 1.0f : 0.0f;   // y[-2]
    float y3 = (j == 2) ? 1.0f : 0.0f;   // y[-3]
    for (int i = 0; i < L; ++i) {
        float y = -(a0 * y1 + a1 * y2 + a2 * y3);
        hp[i * 4 + j] = y;
        y3 = y2; y2 = y1; y1 = y;
    }
}

// ---------------------------------------------------------------------------
// K1: per-(row,chunk) zero-state FIR+AR response z, plus the chunk tail
// (last 3 zero-state outputs) for the inter-chunk state scan.
// One thread per (row, chunk); consecutive threads = consecutive chunks of
// one row (good L2 locality on the streaming reads).
// ---------------------------------------------------------------------------
__global__ __launch_bounds__(256)
void phase1_kernel(const float* __restrict__ in,
                   const float* __restrict__ bco,
                   const float* __restrict__ aco,
                   int nk,
                   float* __restrict__ z,
                   float* __restrict__ tails) {
    const int id = blockIdx.x * blockDim.x + threadIdx.x;   // 0 .. B*C-1
    const int b  = id / C;
    const int c  = id % C;
    const float b0 = bco[0], b1 = bco[1], b2 = bco[2];
    const float a0 = aco[0], a1 = aco[1], a2 = aco[2];

    const float* __restrict__ row = in + (size_t)b * T;
    const int t0 = c * L;
    float x1 = (t0 - nk - 1 >= 0) ? row[t0 - nk - 1] : 0.0f;
    float x2 = (t0 - nk - 2 >= 0) ? row[t0 - nk - 2] : 0.0f;
    float y1 = 0.0f, y2 = 0.0f, y3 = 0.0f;
    float* __restrict__ zrow = z + (size_t)b * T + t0;

#pragma unroll 4
    for (int i = 0; i < L; ++i) {
        const int t = t0 + i;
        if ((i & 63) == 0)  // streaming hint -> global_prefetch_b8
            __builtin_prefetch(row + t + 256, 0, 0);
        float x0 = (t - nk >= 0) ? row[t - nk] : 0.0f;
        float xf = b0 * x0 + b1 * x1 + b2 * x2;
        float y  = xf - (a0 * y1 + a1 * y2 + a2 * y3);
        zrow[i] = y;
        x2 = x1; x1 = x0;
        y3 = y2; y2 = y1; y1 = y;
    }
    float* tl = tails + (size_t)id * 4;
    tl[0] = y1; tl[1] = y2; tl[2] = y3; tl[3] = 0.0f;
}

// ---------------------------------------------------------------------------
// K2: sequential scan over chunk-boundary states (3-vector per row).
// s_{c+1} = ztail_c + Htail * s_c ; Htail rows = h[L-1], h[L-2], h[L-3].
// Stores the state *entering* each chunk as [b*C+c][4] (4th element = 0).
// ---------------------------------------------------------------------------
__global__ __launch_bounds__(256)
void scan_kernel(const float* __restrict__ tails,
                 const float* __restrict__ hpack,
                 float* __restrict__ states) {
    const int b = blockIdx.x * blockDim.x + threadIdx.x;
    if (b >= Bsz) return;
    const float h00 = hpack[(L - 1) * 4 + 0], h01 = hpack[(L - 1) * 4 + 1], h02 = hpack[(L - 1) * 4 + 2];
    const float h10 = hpack[(L - 2) * 4 + 0], h11 = hpack[(L - 2) * 4 + 1], h12 = hpack[(L - 2) * 4 + 2];
    const float h20 = hpack[(L - 3) * 4 + 0], h21 = hpack[(L - 3) * 4 + 1], h22 = hpack[(L - 3) * 4 + 2];
    float s0 = 0.0f, s1 = 0.0f, s2 = 0.0f;
    for (int c = 0; c < C; ++c) {
        const size_t idx = ((size_t)b * C + c) * 4;
        states[idx + 0] = s0; states[idx + 1] = s1;
        states[idx + 2] = s2; states[idx + 3] = 0.0f;
        const float t0 = tails[idx + 0], t1 = tails[idx + 1], t2 = tails[idx + 2];
        const float n0 = t0 + h00 * s0 + h01 * s1 + h02 * s2;
        const float n1 = t1 + h10 * s0 + h11 * s1 + h12 * s2;
        const float n2 = t2 + h20 * s0 + h21 * s1 + h22 * s2;
        s0 = n0; s1 = n1; s2 = n2;
    }
}

// ---------------------------------------------------------------------------
// K3/K6: exact rank-3 correction via V_WMMA_F32_16X16X4_F32.
//   D[m][n] = sum_k H[ib+m][k] * S[k][pair n] + Z[pair n][ib+m]
// Each wave owns 16 (row,chunk) pairs = 16 consecutive chunks of one row.
// A (16x4 f32): lane M = lane&15, K = 2*(lane>>4) + vgpr  -> v2f from hpack.
// B (4x16 f32): lane N = lane&15, K = 2*(lane>>4) + vgpr  -> v2f from states.
// C/D (16x16 f32): vgpr r, lane -> (M = r + 8*(lane>>4), N = lane&15).
// EXEC is all-ones: grid sized exactly, no divergent masking (whf is
// branchless). Final stage stores non-temporally (never re-read on chip).
// ---------------------------------------------------------------------------
template <bool APPLY_NL>
__global__ __launch_bounds__(256)
void correct_kernel(const float* __restrict__ z,
                    const float* __restrict__ states,
                    const float* __restrict__ hpack,
                    float* __restrict__ out) {
    const int lane = threadIdx.x & 31;
    const int wave = threadIdx.x >> 5;
    const int g    = blockIdx.x * (blockDim.x >> 5) + wave;  // pair group
    const int n    = lane & 15;
    const int half = lane >> 4;
    const int b    = g / (C / 16);
    const int cg   = g % (C / 16);
    const int c    = cg * 16 + n;                            // this lane's column/pair

    const float* __restrict__ zb = z   + (size_t)b * T + (size_t)c * L;
    float*       __restrict__ ob = out + (size_t)b * T + (size_t)c * L;

    // B matrix (constant across the whole chunk loop)
    const v2f bv = *(const v2f*)(states + ((size_t)b * C + c) * 4 + 2 * half);

#pragma unroll 4
    for (int ib = 0; ib < L; ib += 16) {
        const v2f av = *(const v2f*)(hpack + (size_t)(ib + n) * 4 + 2 * half);
        const v4f c0 = *(const v4f*)(zb + ib + 8 * half);
        const v4f c1 = *(const v4f*)(zb + ib + 8 * half + 4);
        v8f cv;
        cv[0] = c0[0]; cv[1] = c0[1]; cv[2] = c0[2]; cv[3] = c0[3];
        cv[4] = c1[0]; cv[5] = c1[1]; cv[6] = c1[2]; cv[7] = c1[3];

        v8f dv = __builtin_amdgcn_wmma_f32_16x16x4_f32(
            /*neg_a=*/false, av, /*neg_b=*/false, bv,
            /*c_mod=*/(short)0, cv, /*reuse_a=*/false, /*reuse_b=*/false);

        v4f o0, o1;
        if (APPLY_NL) {
            o0[0] = whf(dv[0]); o0[1] = whf(dv[1]); o0[2] = whf(dv[2]); o0[3] = whf(dv[3]);
            o1[0] = whf(dv[4]); o1[1] = whf(dv[5]); o1[2] = whf(dv[6]); o1[3] = whf(dv[7]);
        } else {
            o0[0] = dv[0]; o0[1] = dv[1]; o0[2] = dv[2]; o0[3] = dv[3];
            o1[0] = dv[4]; o1[1] = dv[5]; o1[2] = dv[6]; o1[3] = dv[7];
        }
        if (APPLY_NL) {
            // v is re-read by stage-2 phase1: keep regular-temporal stores (L2).
            *(v4f*)(ob + ib + 8 * half)     = o0;
            *(v4f*)(ob + ib + 8 * half + 4) = o1;
        } else {
            // final output: never re-read on chip -> non-temporal store
            __builtin_nontemporal_store(o0, (v4f*)(ob + ib + 8 * half));
            __builtin_nontemporal_store(o1, (v4f*)(ob + ib + 8 * half + 4));
        }
    }
}

// ---------------------------------------------------------------------------
// Launch: hpack -> [phase1 -> scan -> correct(+NL into d_out)] for filter 1,
// then the same for filter 2 reading v from d_out and writing final into d_out.
// Workspace: z (B*T) | tails (B*C*4) | states (B*C*4) | hp1 (L*4) | hp2 (L*4)
// ~34 MB total; intermediates stay L2-resident (192 MB).
// ---------------------------------------------------------------------------
extern "C" void kernel_launch(void* const* d_in, const int* in_sizes, int n_in,
                              void* d_out, int out_size, void* d_ws, size_t ws_size,
                              hipStream_t stream) {
    const float* u  = (const float*)d_in[0];
    const float* b1 = (const float*)d_in[1];
    const float* a1 = (const float*)d_in[2];
    const float* b2 = (const float*)d_in[3];
    const float* a2 = (const float*)d_in[4];
    float* out = (float*)d_out;

    float* ws     = (float*)d_ws;
    float* z      = ws;                                   // B*T floats
    float* tails  = z + (size_t)Bsz * T;                  // B*C*4 floats
    float* states = tails + (size_t)Bsz * C * 4;          // B*C*4 floats
    float* hp1    = states + (size_t)Bsz * C * 4;         // L*4 floats
    float* hp2    = hp1 + L * 4;                          // L*4 floats

    const int p1_blocks  = (Bsz * C) / 256;               // 64
    const int sc_blocks  = (Bsz + 255) / 256;             // 2
    const int cor_blocks = (Bsz * (C / 16)) / 8;          // 128 (8 waves/block)

    hpack_kernel<<<2, 32, 0, stream>>>(a1, a2, hp1, hp2);

    // ---- stage 1: y1 = IIR(b1,a1,nk=1)(u), v = whf(y1) -> d_out ----
    phase1_kernel<<<p1_blocks, 256, 0, stream>>>(u, b1, a1, /*nk=*/1, z, tails);
    scan_kernel<<<sc_blocks, 256, 0, stream>>>(tails, hp1, states);
    correct_kernel<true><<<cor_blocks, 256, 0, stream>>>(z, states, hp1, out);

    // ---- stage 2: y2 = IIR(b2,a2,nk=0)(v) -> d_out (overwrites v) ----
    phase1_kernel<<<p1_blocks, 256, 0, stream>>>(out, b2, a2, /*nk=*/0, z, tails);
    scan_kernel<<<sc_blocks, 256, 0, stream>>>(tails, hp2, states);
    correct_kernel<false><<<cor_blocks, 256, 0, stream>>>(z, states, hp2, out);
}